// QuantumLayer_32100585571044
// MI455X (gfx1250) — compile-verified
//
#include <hip/hip_runtime.h>

#define NQ 4
#define DEPTH 6

typedef float v2f __attribute__((ext_vector_type(2)));
typedef float v8f __attribute__((ext_vector_type(8)));

// Swizzled W layout: float2 element index = (k*2 + h)*32 + L, component v.
//   m = 4k + K  (K = v + 2*(L>=16)),  column = 16h + (L&15)
//   columns 0..15 = Re(amp_i), 16..31 = Im(amp_i)
__device__ __forceinline__ int wswz_idx(int m, int col) {
  int k = m >> 2, K = m & 3;
  int h = col >> 4;
  int L = (col & 15) + 16 * (K >> 1);
  return (((k * 2 + h) * 32) + L) * 2 + (K & 1);
}

// ---------------- kernel 0: zero the 32KB W buffer ----------------
__global__ void zeroW_kern(float* __restrict__ w) {
  int i = blockIdx.x * blockDim.x + threadIdx.x;
  if (i < 8192) w[i] = 0.0f;
}

// ---------------- kernel 1: build W from 4096 circuit variants ----------------
__device__ __forceinline__ void mul1q(float* re, float* im, int w,
                                      float u00r, float u00i, float u01r, float u01i,
                                      float u10r, float u10i, float u11r, float u11i) {
  const int m = 8 >> w;
#pragma unroll
  for (int i = 0; i < 16; ++i) {
    if ((i & m) == 0) {
      int j = i | m;
      float ar = re[i], ai = im[i], br = re[j], bi = im[j];
      re[i] = u00r * ar - u00i * ai + u01r * br - u01i * bi;
      im[i] = u00r * ai + u00i * ar + u01r * bi + u01i * br;
      re[j] = u10r * ar - u10i * ai + u11r * br - u11i * bi;
      im[j] = u10r * ai + u10i * ar + u11r * bi + u11i * br;
    }
  }
}

__global__ void buildW_kern(const float* __restrict__ qp, float* __restrict__ wout) {
  int tid = blockIdx.x * blockDim.x + threadIdx.x;
  if (tid >= 4096) return;

  const int mask0 = tid & 7, mask1 = (tid >> 3) & 7, mask2 = (tid >> 6) & 7, mask3 = (tid >> 9) & 7;
  const int masks[4] = {mask0, mask1, mask2, mask3};

  float re[16], im[16];
#pragma unroll
  for (int i = 0; i < 16; ++i) { re[i] = 0.25f; im[i] = 0.0f; }

  const float SQ2 = 0.70710678118654752440f;

#pragma unroll
  for (int l = 0; l < DEPTH; ++l) {
#pragma unroll
    for (int n = 0; n < NQ; ++n) {
      const bool use_in = (((l + n) & 1) == 0);
      const int rot = (l + n) % 3;          // 0=x 1=y 2=z
      const int extra = (l * NQ + n) & 7;   // 0 CNot,1 H,2 CZ,3 Tof,4 Pz,5 CSwap,6 Px,7 Py

      if (use_in) {
        // occurrence index on wire n is l>>1 (3 occurrences per wire)
        if ((masks[n] >> (l >> 1)) & 1) {
          // G = [[0,-1],[1,0]]   (RY = c*I + s*G)
          mul1q(re, im, n, 0.f, 0.f, -1.f, 0.f, 1.f, 0.f, 0.f, 0.f);
        }
        // else: identity (the c-branch), nothing to do
      }

      {
        float a = qp[l * NQ + n];
        float c = cosf(0.5f * a), s = sinf(0.5f * a);
        if (rot == 0) {        // RX: [[c,-is],[-is,c]]
          mul1q(re, im, n, c, 0.f, 0.f, -s, 0.f, -s, c, 0.f);
        } else if (rot == 1) { // RY: [[c,-s],[s,c]]
          mul1q(re, im, n, c, 0.f, -s, 0.f, s, 0.f, c, 0.f);
        } else {               // RZ: diag(c-is, c+is)
          mul1q(re, im, n, c, -s, 0.f, 0.f, 0.f, 0.f, c, s);
        }
      }

      if (extra == 1) {        // H
        mul1q(re, im, n, SQ2, 0.f, SQ2, 0.f, SQ2, 0.f, -SQ2, 0.f);
      } else if (extra == 6) { // Px
        mul1q(re, im, n, 0.f, 0.f, 1.f, 0.f, 1.f, 0.f, 0.f, 0.f);
      } else if (extra == 7) { // Py
        mul1q(re, im, n, 0.f, 0.f, 0.f, -1.f, 0.f, 1.f, 0.f, 0.f);
      } else if (extra == 4) { // Pz
        mul1q(re, im, n, 1.f, 0.f, 0.f, 0.f, 0.f, 0.f, -1.f, 0.f);
      } else if (extra == 0) { // CNot(n, n+1)
        const int mc = 8 >> n, mt = 8 >> ((n + 1) & 3);
#pragma unroll
        for (int i = 0; i < 16; ++i) {
          if ((i & mc) && !(i & mt)) {
            int j = i | mt;
            float tr = re[i]; re[i] = re[j]; re[j] = tr;
            float ti = im[i]; im[i] = im[j]; im[j] = ti;
          }
        }
      } else if (extra == 2) { // CZ(n, n+1)
        const int mc = 8 >> n, mt = 8 >> ((n + 1) & 3);
#pragma unroll
        for (int i = 0; i < 16; ++i) {
          if ((i & mc) && (i & mt)) { re[i] = -re[i]; im[i] = -im[i]; }
        }
      } else if (extra == 3) { // Toffoli(n, n+1, n+2)
        const int ma = 8 >> n, mb = 8 >> ((n + 1) & 3), mc2 = 8 >> ((n + 2) & 3);
#pragma unroll
        for (int i = 0; i < 16; ++i) {
          if ((i & ma) && (i & mb) && !(i & mc2)) {
            int j = i | mc2;
            float tr = re[i]; re[i] = re[j]; re[j] = tr;
            float ti = im[i]; im[i] = im[j]; im[j] = ti;
          }
        }
      } else if (extra == 5) { // CSwap(n, n+1, n+2)
        const int ma = 8 >> n, mb = 8 >> ((n + 1) & 3), mc2 = 8 >> ((n + 2) & 3);
#pragma unroll
        for (int i = 0; i < 16; ++i) {
          if ((i & ma) && (i & mb) && !(i & mc2)) {
            int j = (i ^ mb) | mc2;
            float tr = re[i]; re[i] = re[j]; re[j] = tr;
            float ti = im[i]; im[i] = im[j]; im[j] = ti;
          }
        }
      }
    }
  }

  // bin index from per-wire popcounts
  const int m = __popc(mask0) + 4 * __popc(mask1) + 16 * __popc(mask2) + 64 * __popc(mask3);
#pragma unroll
  for (int i = 0; i < 16; ++i) {
    atomicAdd(&wout[wswz_idx(m, i)], re[i]);
    atomicAdd(&wout[wswz_idx(m, 16 + i)], im[i]);
  }
}

// ---------------- kernel 2: monomials + f32 WMMA GEMM + <Z> reduction ----------------
__global__ void __launch_bounds__(128) qgemm_kern(const float* __restrict__ feat,
                                                  const v2f* __restrict__ Wz,
                                                  float* __restrict__ out) {
  const int lane = threadIdx.x & 31;
  const int wid = blockIdx.x * (blockDim.x >> 5) + (threadIdx.x >> 5);
  const int sbase = wid << 4;         // 16 samples per wave
  const int srow = sbase + (lane & 15);
  const int half = lane >> 4;

  const float4 f = ((const float4*)feat)[srow];
  float c0 = cosf(0.5f * f.x), s0 = sinf(0.5f * f.x);
  float c1 = cosf(0.5f * f.y), s1 = sinf(0.5f * f.y);
  float c2 = cosf(0.5f * f.z), s2 = sinf(0.5f * f.z);
  float c3 = cosf(0.5f * f.w), s3 = sinf(0.5f * f.w);

  // p_n[j] = c^(3-j) s^j
  float P0[4], P1[4], P2[4], P3[4];
  {
    float cc = c0 * c0, ss = s0 * s0;
    P0[0] = cc * c0; P0[1] = cc * s0; P0[2] = c0 * ss; P0[3] = ss * s0;
  }
  {
    float cc = c1 * c1, ss = s1 * s1;
    P1[0] = cc * c1; P1[1] = cc * s1; P1[2] = c1 * ss; P1[3] = ss * s1;
  }
  {
    float cc = c2 * c2, ss = s2 * s2;
    P2[0] = cc * c2; P2[1] = cc * s2; P2[2] = c2 * ss; P2[3] = ss * s2;
  }
  {
    float cc = c3 * c3, ss = s3 * s3;
    P3[0] = cc * c3; P3[1] = cc * s3; P3[2] = c3 * ss; P3[3] = ss * s3;
  }

  // A-operand K rows this lane supplies: K = {2*half, 2*half+1}
  const float pa = half ? P0[2] : P0[0];
  const float pb = half ? P0[3] : P0[1];

  v8f acc0 = {0.f, 0.f, 0.f, 0.f, 0.f, 0.f, 0.f, 0.f}; // Re columns
  v8f acc1 = {0.f, 0.f, 0.f, 0.f, 0.f, 0.f, 0.f, 0.f}; // Im columns

#pragma unroll
  for (int k = 0; k < 64; ++k) {
    float q = P1[k & 3] * P2[(k >> 2) & 3] * P3[(k >> 4) & 3];
    v2f A = {pa * q, pb * q};
    v2f b0 = Wz[(2 * k + 0) * 32 + lane];
    v2f b1 = Wz[(2 * k + 1) * 32 + lane];
    acc0 = __builtin_amdgcn_wmma_f32_16x16x4_f32(false, A, false, b0, (short)0, acc0, false, false);
    acc1 = __builtin_amdgcn_wmma_f32_16x16x4_f32(false, A, false, b1, (short)0, acc1, false, false);
  }

  // D layout: lane holds (M = r + 8*half, N = lane&15). Reduce over N within 16-lane group.
  const int g = lane & 15;
  const float sg1 = (lane & 2) ? -1.f : 1.f; // qubit 2 (bit1 of i)  -> output p=2
  const float sg2 = (lane & 4) ? -1.f : 1.f; // qubit 1 (bit2)      -> output p=1
  const float sg3 = (lane & 8) ? -1.f : 1.f; // qubit 0 (bit3)      -> output p=0

#pragma unroll
  for (int r = 0; r < 8; ++r) {
    float Re = acc0[r], Im = acc1[r];
    float P = Re * Re + Im * Im;
    float Q = P + __shfl_xor(P, 1, 32);      // sum over bit0 (qubit 3, unused)

    float T2 = sg1 * Q;
    T2 += __shfl_xor(T2, 2, 32);
    T2 += __shfl_xor(T2, 4, 32);
    T2 += __shfl_xor(T2, 8, 32);

    float Q2 = Q + __shfl_xor(Q, 2, 32);     // also sum over bit1
    float T1 = sg2 * Q2;
    T1 += __shfl_xor(T1, 4, 32);
    T1 += __shfl_xor(T1, 8, 32);

    float T0 = sg3 * Q2;
    T0 += __shfl_xor(T0, 4, 32);
    T0 += __shfl_xor(T0, 8, 32);

    if (g < 3) {
      float e = (g == 0) ? T0 : (g == 1) ? T1 : T2;
      out[(sbase + r + 8 * half) * 3 + g] = e;
    }
  }
}

extern "C" void kernel_launch(void* const* d_in, const int* in_sizes, int n_in,
                              void* d_out, int out_size, void* d_ws, size_t ws_size,
                              hipStream_t stream) {
  const float* feat = (const float*)d_in[0];   // [B,4] f32
  const float* qp = (const float*)d_in[1];     // [24]  f32
  float* W = (float*)d_ws;                     // 8192 f32 = 32KB swizzled W
  float* out = (float*)d_out;                  // [B,3] f32

  const int B = in_sizes[0] / 4;               // 131072
  const int waves = B / 16;                    // 8192
  const int blocks = waves / 4;                // 4 waves (128 threads) per block

  zeroW_kern<<<32, 256, 0, stream>>>(W);
  buildW_kern<<<32, 128, 0, stream>>>(qp, W);
  qgemm_kern<<<blocks, 128, 0, stream>>>(feat, (const v2f*)W, out);
}